// STGCNEncoder_22471268893029
// MI455X (gfx1250) — compile-verified
//
#include <hip/hip_runtime.h>
#include <hip/hip_bf16.h>

// ---------------------------------------------------------------------------
// STGCN encoder, pruned to the live data flow:
// output = GRU(g[0], h) where g[0] depends only on edges with dst==0.
//
// Pipeline: memset -> edge_scan (b128 streaming + L2 atomics)
//        -> gemm_rows (f32 WMMA 16x16x4 on the ~33 live rows)
//        -> agent (1 block)  -> gru_matvec (192 blocks, coalesced, wave-reduce)
//        -> gru_gates (1 block)
// ---------------------------------------------------------------------------

typedef __attribute__((ext_vector_type(2))) float v2f;
typedef __attribute__((ext_vector_type(8))) float v8f;

#define N_NODES 100000
#define E_EDGES 3200000
#define CAP     4096          // capacity for edges hitting node 0 (E/N ~ 32 expected)

// workspace layout (bytes)
#define OFF_DEG   0u                   // N_NODES ints
#define OFF_CNT   400000u              // 1 int  (== N_NODES*4)
#define OFF_LIST  400128u              // CAP ints
#define OFF_XWC   416640u              // (CAP+16) rows x 128 floats = 2,105,344 B
#define OFF_AGENT 2521984u             // 128 floats
#define OFF_GIH   2522496u             // 1536 floats (gi[768] ++ gh[768])

// ---------------------------------------------------------------------------
// Pass 1: degree histogram over dst; collect srcs of edges with dst==0.
// 4 edges per thread via b128 loads -> 512B per wave per issue, HBM-limited.
// ---------------------------------------------------------------------------
__global__ __launch_bounds__(256) void edge_scan_kernel(
    const int* __restrict__ ei, int* __restrict__ deg,
    int* __restrict__ counter, int* __restrict__ list)
{
  const int t = blockIdx.x * 256 + threadIdx.x;   // t < E/4
  const int4 d = ((const int4*)(ei + E_EDGES))[t];

  atomicAdd(&deg[d.x], 1);
  atomicAdd(&deg[d.y], 1);
  atomicAdd(&deg[d.z], 1);
  atomicAdd(&deg[d.w], 1);

  const int e0 = 4 * t;
  if (d.x == 0) { int p = atomicAdd(counter, 1); if (p < CAP) list[p] = ei[e0 + 0]; }
  if (d.y == 0) { int p = atomicAdd(counter, 1); if (p < CAP) list[p] = ei[e0 + 1]; }
  if (d.z == 0) { int p = atomicAdd(counter, 1); if (p < CAP) list[p] = ei[e0 + 2]; }
  if (d.w == 0) { int p = atomicAdd(counter, 1); if (p < CAP) list[p] = ei[e0 + 3]; }
}

// ---------------------------------------------------------------------------
// Pass 2: xw = relu(nf @ W_enc^T + b_enc) @ W_gcn^T for the compacted row set.
// Slot 0 = node 0 (self loop), slots 1..cnt = list[i-1].
// One block = 16-row tile, 8 waves, wave w owns output columns [16w,16w+16).
// f32 WMMA 16x16x4; operand packing per CDNA5 ISA 7.12.2 (32-bit layouts):
//   A: lane&15 = M, lane>>4 selects K-pair {0,1}/{2,3} of each K4 step
//   B: lane&15 = N, same K-pair split
//   C/D: VGPR i -> row i (lanes 0-15) / i+8 (lanes 16-31), col = lane&15
// ---------------------------------------------------------------------------
__global__ __launch_bounds__(256) void gemm_rows_kernel(
    const float* __restrict__ nf,   const float* __restrict__ Wenc,
    const float* __restrict__ benc, const float* __restrict__ Wgcn,
    const int*   __restrict__ counter, const int* __restrict__ list,
    float* __restrict__ xwc)
{
  __shared__ float xs[16][132];     // padded: stride 132 words -> conflict-free
  __shared__ int   rows[16];
  __shared__ int   s_total;

  const int tid = threadIdx.x;
  if (tid == 0) {
    int c = *counter; if (c > CAP) c = CAP;
    s_total = c + 1;                // +1 for node-0 slot
  }
  __syncthreads();
  const int total = s_total;
  if ((int)blockIdx.x * 16 >= total) return;   // uniform early exit

  if (tid < 16) {
    int gi = blockIdx.x * 16 + tid;
    int r = 0;                                  // pad rows replicate node 0
    if (gi > 0 && gi < total) r = list[gi - 1];
    rows[tid] = r;
  }
  __syncthreads();

  const int lane = tid & 31;
  const int half = lane >> 4;       // 0: K={0,1}, 1: K={2,3}
  const int m    = lane & 15;       // M for A, N for B/C/D
  const int koff = half * 2;
  const int n0   = (tid >> 5) * 16; // wave's output-column base

  // ---- GEMM1: x chunk = nf_rows(16x64) * W_enc^T ----
  v8f acc = {0.f,0.f,0.f,0.f,0.f,0.f,0.f,0.f};
  {
    const float* Arow = nf   + (size_t)rows[m] * 64;       // A[m][k]
    const float* Brow = Wenc + (size_t)(n0 + m) * 64;      // B[k][n]=W_enc[n][k]
#pragma unroll
    for (int s = 0; s < 16; ++s) {
      const int k = s * 4 + koff;
      v2f a = { Arow[k], Arow[k + 1] };
      v2f b = { Brow[k], Brow[k + 1] };
      acc = __builtin_amdgcn_wmma_f32_16x16x4_f32(
                false, a, false, b, (short)0, acc, false, false);
    }
  }
  {
    const float bias = benc[n0 + m];
#pragma unroll
    for (int i = 0; i < 8; ++i) {
      const int r = i + half * 8;
      float v = acc[i] + bias;
      xs[r][n0 + m] = v > 0.f ? v : 0.f;       // relu
    }
  }
  __syncthreads();

  // ---- GEMM2: xw chunk = x_tile(16x128) * W_gcn^T ----
  v8f acc2 = {0.f,0.f,0.f,0.f,0.f,0.f,0.f,0.f};
  {
    const float* Brow = Wgcn + (size_t)(n0 + m) * 128;     // B[k][n]=W_gcn[n][k]
#pragma unroll
    for (int s = 0; s < 32; ++s) {
      const int k = s * 4 + koff;
      v2f a = { xs[m][k], xs[m][k + 1] };
      v2f b = { Brow[k], Brow[k + 1] };
      acc2 = __builtin_amdgcn_wmma_f32_16x16x4_f32(
                 false, a, false, b, (short)0, acc2, false, false);
    }
  }
#pragma unroll
  for (int i = 0; i < 8; ++i) {
    const int r = i + half * 8;
    xwc[(size_t)(blockIdx.x * 16 + r) * 128 + n0 + m] = acc2[i];
  }
}

// ---------------------------------------------------------------------------
// Pass 3a (1 block, 128 threads): agent = relu(agg[0] + b_gcn)
// ---------------------------------------------------------------------------
__global__ __launch_bounds__(128) void agent_kernel(
    const float* __restrict__ xwc,  const int* __restrict__ deg,
    const int*   __restrict__ counter, const int* __restrict__ list,
    const float* __restrict__ bgcn, float* __restrict__ agent)
{
  const int tid = threadIdx.x;        // channel 0..127
  int cnt = *counter; if (cnt > CAP) cnt = CAP;
  const float dinv0 = rsqrtf((float)deg[0] + 1.0f);

  float acc = dinv0 * dinv0 * xwc[tid];            // self loop (slot 0 = node 0)
  for (int i = 0; i < cnt; ++i) {
    const int s = list[i];
    const float nrm = dinv0 * rsqrtf((float)deg[s] + 1.0f);
    acc += nrm * xwc[(size_t)(i + 1) * 128 + tid];
  }
  float v = acc + bgcn[tid];
  agent[tid] = v > 0.f ? v : 0.f;
}

// ---------------------------------------------------------------------------
// Pass 3b (192 blocks x 8 waves): one wave per output row, coalesced K-stride,
// wave32 xor-shuffle reduction. Rows 0..767 -> gi, rows 768..1535 -> gh.
// ---------------------------------------------------------------------------
__global__ __launch_bounds__(256) void gru_matvec_kernel(
    const float* __restrict__ agent, const float* __restrict__ h,
    const float* __restrict__ Wih,   const float* __restrict__ bih,
    const float* __restrict__ Whh,   const float* __restrict__ bhh,
    float* __restrict__ gih)
{
  const int lane = threadIdx.x & 31;
  const int row  = blockIdx.x * 8 + (threadIdx.x >> 5);   // 0..1535

  float acc = 0.f;
  float bias;
  if (row < 768) {                       // gi[row] = agent . W_ih[row]
    const float* w = Wih + (size_t)row * 128;
#pragma unroll
    for (int i = 0; i < 4; ++i) {
      const int k = lane + 32 * i;
      acc += w[k] * agent[k];
    }
    bias = bih[row];
  } else {                               // gh[j] = h . W_hh[j]
    const int j = row - 768;
    const float* w = Whh + (size_t)j * 256;
#pragma unroll
    for (int i = 0; i < 8; ++i) {
      const int k = lane + 32 * i;
      acc += w[k] * h[k];
    }
    bias = bhh[j];
  }
#pragma unroll
  for (int m = 16; m >= 1; m >>= 1)
    acc += __shfl_xor(acc, m, 32);
  if (lane == 0) gih[row] = acc + bias;
}

// ---------------------------------------------------------------------------
// Pass 3c (1 block, 256 threads): GRU gates -> out(256).
// ---------------------------------------------------------------------------
__global__ __launch_bounds__(256) void gru_gates_kernel(
    const float* __restrict__ gih, const float* __restrict__ h,
    float* __restrict__ out)
{
  const int j = threadIdx.x;
  const float* gi = gih;
  const float* gh = gih + 768;
  const float r  = 1.f / (1.f + __expf(-(gi[j]       + gh[j])));
  const float z  = 1.f / (1.f + __expf(-(gi[256 + j] + gh[256 + j])));
  const float nn = tanhf(gi[512 + j] + r * gh[512 + j]);
  out[j] = (1.f - z) * nn + z * h[j];
}

// ---------------------------------------------------------------------------
extern "C" void kernel_launch(void* const* d_in, const int* in_sizes, int n_in,
                              void* d_out, int out_size, void* d_ws, size_t ws_size,
                              hipStream_t stream)
{
  (void)in_sizes; (void)n_in; (void)out_size; (void)ws_size;

  const float* nf   = (const float*)d_in[0];   // node_features (N,64)
  // d_in[1] edge_attr: unused by the reference
  const float* h    = (const float*)d_in[2];   // hidden_state (1,256)
  const float* Wenc = (const float*)d_in[3];   // (128,64)
  const float* benc = (const float*)d_in[4];   // (128,)
  const float* Wgcn = (const float*)d_in[5];   // (128,128)
  const float* bgcn = (const float*)d_in[6];   // (128,)
  const float* Wih  = (const float*)d_in[7];   // (768,128)
  const float* Whh  = (const float*)d_in[8];   // (768,256)
  const float* bih  = (const float*)d_in[9];   // (768,)
  const float* bhh  = (const float*)d_in[10];  // (768,)
  const int*   ei   = (const int*)d_in[11];    // edge_index (2,E)

  char* ws = (char*)d_ws;
  int*   deg     = (int*)(ws + OFF_DEG);
  int*   counter = (int*)(ws + OFF_CNT);
  int*   list    = (int*)(ws + OFF_LIST);
  float* xwc     = (float*)(ws + OFF_XWC);
  float* agent   = (float*)(ws + OFF_AGENT);
  float* gih     = (float*)(ws + OFF_GIH);

  // zero degree histogram + edge counter (graph-capturable)
  hipMemsetAsync(ws, 0, (size_t)N_NODES * 4 + 64, stream);

  edge_scan_kernel<<<E_EDGES / 4 / 256, 256, 0, stream>>>(ei, deg, counter, list);
  gemm_rows_kernel<<<CAP / 16, 256, 0, stream>>>(nf, Wenc, benc, Wgcn, counter, list, xwc);
  agent_kernel<<<1, 128, 0, stream>>>(xwc, deg, counter, list, bgcn, agent);
  gru_matvec_kernel<<<192, 256, 0, stream>>>(agent, h, Wih, bih, Whh, bhh, gih);
  gru_gates_kernel<<<1, 256, 0, stream>>>(gih, h, (float*)d_out);
}